// DCTLayer_46119358824462
// MI455X (gfx1250) — compile-verified
//
#include <hip/hip_runtime.h>

// ---------------------------------------------------------------------------
// DCT 8x8 layer for MI455X (gfx1250), fully folded:
//   out_block = (D*D*X)^T  ==  X^T * (D*D)^T        (per 8x8 block)
// Memory-bound: 536 MB traffic => ~23 us floor @ 23.3 TB/s. Math is done with
// V_WMMA_F32_16X16X4_F32: constant B = blockdiag(M^T,M^T), streamed A = 16x16
// pixel tile with quadrant-transpose, so C-fragments are already in output
// memory layout (no post-shuffle).
// ---------------------------------------------------------------------------

typedef float v2f __attribute__((ext_vector_type(2)));
typedef float v4f __attribute__((ext_vector_type(4)));
typedef float v8f __attribute__((ext_vector_type(8)));

#define IMG_W 256
#define PITCH 36 // LDS row pitch (floats): 16B-aligned rows, conflict-free A reads

__global__ __launch_bounds__(256, 2) void dct8x8_wmma_kernel(
    const float* __restrict__ x, const float* __restrict__ dct,
    float* __restrict__ out)
{
    __shared__ float sM[64];              // M = D*D (8x8)
    __shared__ float stage[8][16 * PITCH]; // per-wave 16x32 input staging (+pad)

    const int tid  = threadIdx.x;
    const int lane = tid & 31;
    const int wave = tid >> 5;

    // ---- M = D*D, computed once per block (trivial: 64 threads x 8 FMA) ----
    if (tid < 64) {
        const int i = tid >> 3, j = tid & 7;
        float acc = 0.f;
#pragma unroll
        for (int t = 0; t < 8; ++t)
            acc += dct[i * 8 + t] * dct[t * 8 + j];
        sM[tid] = acc;
    }
    __syncthreads();

    // ---- constant B fragments: 16x16 blockdiag(M^T, M^T), K-chunks of 4 ----
    // B layout guess (mirrors documented A/C layouts): VGPR v of a 4x16 chunk,
    // lane l holds B[4k + v + 2*(l>>4)][l&15].
    const int jhi = (lane >> 4) << 1;
    v2f bfrag[4];
#pragma unroll
    for (int k = 0; k < 4; ++k) {
#pragma unroll
        for (int v = 0; v < 2; ++v) {
            const int row = 4 * k + v + jhi;
            const int col = lane & 15;
            // M^T[row%8][col%8] on the diagonal quadrants, else 0
            bfrag[k][v] = (((row ^ col) & 8) == 0)
                              ? sM[(col & 7) * 8 + (row & 7)]
                              : 0.f;
        }
    }

    // ---- supertile (16 rows x 32 cols) assignment: one per wave ----
    const int st   = blockIdx.x * 8 + wave;      // 131072 supertiles total
    const int img  = st >> 7;                    // 16*64 = 1024 images
    const int rem  = st & 127;                   // 16 x 8 supertiles / image
    const int h0   = (rem >> 3) << 4;
    const int w0   = (rem & 7) << 5;
    const float* __restrict__ src = x   + (size_t)img * (IMG_W * IMG_W);
    float*       __restrict__ dst = out + (size_t)img * (IMG_W * IMG_W);
    float* lds = &stage[wave][0];

    // ---- global -> LDS: 4x global_load_b128, fully 128B-coalesced, NT ----
    {
        const int r  = lane >> 3;        // 0..3
        const int c4 = (lane & 7) << 2;  // 0,4,...,28
#pragma unroll
        for (int it = 0; it < 4; ++it) {
            const int row = it * 4 + r;
            v4f d = __builtin_nontemporal_load(
                (const v4f*)(src + (size_t)(h0 + row) * IMG_W + w0 + c4));
            *(v4f*)(lds + row * PITCH + c4) = d;
        }
    }
    // wave-local LDS RAW fence (DS ops in-order within a wave; each wave owns
    // its own staging slice, so no workgroup barrier needed)
    asm volatile("s_wait_dscnt 0" ::: "memory");

    // ---- per 16x16 tile: A = quadrant-transposed tile, 4x WMMA K-chunks ----
    const int i = lane & 15; // A row index held by this lane
#pragma unroll
    for (int tc = 0; tc < 2; ++tc) {
        v8f c = {};
#pragma unroll
        for (int k = 0; k < 4; ++k) {
            v2f a;
#pragma unroll
            for (int v = 0; v < 2; ++v) {
                const int j = 4 * k + v + jhi; // A col (K index)
                // A[i][j] = tile[(i&8)|(j&7)][(j&8)|(i&7)]  (quadrant transpose)
                a[v] = lds[((i & 8) + (j & 7)) * PITCH + tc * 16 + (j & 8) + (i & 7)];
            }
            c = __builtin_amdgcn_wmma_f32_16x16x4_f32(
                /*neg_a=*/false, a, /*neg_b=*/false, bfrag[k],
                /*c_mod=*/(short)0, c, /*reuse_a=*/false, /*reuse_b=*/false);
        }
        // ---- C fragment is already in output layout: direct NT stores ----
        const int col   = lane & 15;
        const int rbase = (lane >> 4) << 3;
        float* dcol = dst + (size_t)(h0 + rbase) * IMG_W + w0 + tc * 16 + col;
#pragma unroll
        for (int r = 0; r < 8; ++r)
            __builtin_nontemporal_store(c[r], dcol + (size_t)r * IMG_W);
    }
}

extern "C" void kernel_launch(void* const* d_in, const int* in_sizes, int n_in,
                              void* d_out, int out_size, void* d_ws, size_t ws_size,
                              hipStream_t stream)
{
    (void)in_sizes; (void)n_in; (void)out_size; (void)d_ws; (void)ws_size;
    const float* x   = (const float*)d_in[0]; // (16,64,256,256) f32
    const float* dct = (const float*)d_in[1]; // (8,8) f32
    float*       out = (float*)d_out;         // (16,64,256,256) f32

    // 16*64*256*256 elems / (8 waves * 16*32 elems) = 16384 blocks of 256 thr
    dct8x8_wmma_kernel<<<16384, 256, 0, stream>>>(x, dct, out);
}